// SignalGenerator_81003083203024
// MI455X (gfx1250) — compile-verified
//
#include <hip/hip_runtime.h>

// Pink-noise 6-pole IIR, parallelized as a hierarchical linear scan with the
// intra-tile part expressed as 16x16 lower-triangular matmuls on the CDNA5
// WMMA f32 pipe (V_WMMA_F32_16X16X4_F32 x4 per 16x16x16 product).

#define L_TOTAL   65536
#define CHUNK     4096                // samples per scan chunk (one wave)
#define K_PER_CH  (L_TOTAL / CHUNK)   // 16 chunks per channel
#define BLK_SAMP  256                 // samples per WMMA block (16x16 tile)
#define NBLK      (CHUNK / BLK_SAMP)  // 16 blocks per chunk

typedef float v2f __attribute__((ext_vector_type(2)));
typedef float v8f __attribute__((ext_vector_type(8)));

__device__ __constant__ float PA[6] = {0.99886f, 0.99332f, 0.969f, 0.8665f, 0.55f, -0.7616f};
__device__ __constant__ float PC[6] = {0.0555179f, 0.0750759f, 0.153852f, 0.3104856f, 0.5329522f, -0.016898f};
#define B6_GAIN  0.115926f
#define DIRECT_G 0.5362f
#define OUT_GAIN 0.11f

__device__ __forceinline__ float powi(float a, int n) {
  float r = 1.f;
  for (int i = 0; i < n; ++i) r *= a;
  return r;
}
// h[d] = sum_i c_i * a_i^d  (entries of the lower-triangular P matrix)
__device__ __forceinline__ float hsum_pow(int d) {
  float s = 0.f;
#pragma unroll
  for (int i = 0; i < 6; ++i) s += PC[i] * powi(PA[i], d);
  return s;
}

// ---------------------------------------------------------------------------
// Kernel 1: per-chunk zero-init final pole states
//   f_i(chunk) = c_i * sum_{s=0}^{N-1} a_i^{N-1-s} * w[s]
// One wave per chunk; lane l handles samples s = 32j + l (coalesced).
// ---------------------------------------------------------------------------
__global__ void __launch_bounds__(256)
k1_chunk_reduce(const float* __restrict__ w, float* __restrict__ f, int nchunks) {
  const int lane = threadIdx.x & 31;
  const int wave = (int)((blockIdx.x * blockDim.x + threadIdx.x) >> 5);
  if (wave >= nchunks) return;
  const long long base = (long long)wave * CHUNK;

  float r32[6], p[6];
#pragma unroll
  for (int i = 0; i < 6; ++i) { r32[i] = powi(PA[i], 32); p[i] = 0.f; }

  for (int j = 0; j < CHUNK / 32; ++j) {
    float x = w[base + (long long)j * 32 + lane];
#pragma unroll
    for (int i = 0; i < 6; ++i) p[i] = fmaf(r32[i], p[i], x);  // sum r32^{J-1-j} x_j
  }
#pragma unroll
  for (int i = 0; i < 6; ++i) {
    float m = p[i] * powi(PA[i], 31 - lane);                   // a^{31-l} weighting
#pragma unroll
    for (int s = 1; s < 32; s <<= 1) m += __shfl_xor(m, s, 32);
    if (lane == 0) f[(long long)wave * 6 + i] = PC[i] * m;
  }
}

// ---------------------------------------------------------------------------
// Kernel 2: per-channel scan over chunks (16 steps): chunk-entry states.
// ---------------------------------------------------------------------------
__global__ void __launch_bounds__(256)
k2_scan(const float* __restrict__ f, float* __restrict__ st, int nchan) {
  int ch = blockIdx.x * blockDim.x + threadIdx.x;
  if (ch >= nchan) return;
  float rN[6], s[6];
#pragma unroll
  for (int i = 0; i < 6; ++i) {
    float x = PA[i];
    for (int q = 0; q < 12; ++q) x *= x;   // a^(2^12) = a^4096
    rN[i] = x; s[i] = 0.f;
  }
  for (int k = 0; k < K_PER_CH; ++k) {
    long long c = (long long)ch * K_PER_CH + k;
#pragma unroll
    for (int i = 0; i < 6; ++i) {
      st[c * 6 + i] = s[i];                       // state ENTERING chunk k
      s[i] = fmaf(rN[i], s[i], f[c * 6 + i]);
    }
  }
}

// ---------------------------------------------------------------------------
// Kernel 3: main pass. One wave per chunk; per 256-sample block:
//   acc = P @ W        (4 chained V_WMMA_F32_16X16X4_F32)
//   d_i[g] = c_i * q_i . W[:,g]   (VALU + cross-half shfl_xor)
//   decaying Hillis-Steele scan across 16 columns -> per-column carry states
//   out = 0.11*(acc + carry + 0.115926*w[t-1] + 0.5362*w[t])
// ---------------------------------------------------------------------------
__global__ void __launch_bounds__(256)
k3_pink(const float* __restrict__ w, const float* __restrict__ st,
        float* __restrict__ out, int nchunks) {
  const int lane = threadIdx.x & 31;
  const int wave = (int)((blockIdx.x * blockDim.x + threadIdx.x) >> 5);
  if (wave >= nchunks) return;
  const int half = lane >> 4;        // lane half selects K/M sub-rows
  const int col  = lane & 15;        // tile column (N) owned by this lane
  const int m    = col;              // output row for A-matrix build
  const long long chunkBase = (long long)wave * CHUNK;
  const long long chanStart = (chunkBase / L_TOTAL) * L_TOTAL;

  // ---- A matrix (P), WMMA 16x4 f32 layout: lane<16 holds K={4j,4j+1},
  //      lane>=16 holds K={4j+2,4j+3}, row M = lane&15. P[m,k]=h[m-k] (k<=m).
  v2f Amat[4];
#pragma unroll
  for (int j = 0; j < 4; ++j) {
    int k0 = 4 * j + 2 * half;
    Amat[j].x = (k0     <= m) ? hsum_pow(m - k0)     : 0.f;
    Amat[j].y = (k0 + 1 <= m) ? hsum_pow(m - k0 - 1) : 0.f;
  }
  // ---- q coefficients for column dots: cq[i][p] = c_i * a_i^{15-row(p)},
  //      row(p) = 4*(p/2) + (p%2) + 2*half  (the 8 W rows this lane holds)
  float cq[6][8];
#pragma unroll
  for (int p = 0; p < 8; ++p) {
    int row = 4 * (p >> 1) + (p & 1) + 2 * half;
#pragma unroll
    for (int i = 0; i < 6; ++i) cq[i][p] = PC[i] * powi(PA[i], 15 - row);
  }
  // ---- decay powers: r = a^16 (per-column step), rg = r^col, api = a^{1+8h}
  float r1[6], r2[6], r4[6], r8[6], r16[6], rg[6], api[6];
#pragma unroll
  for (int i = 0; i < 6; ++i) {
    float r = powi(PA[i], 16);
    r1[i] = r; r2[i] = r * r; r4[i] = r2[i] * r2[i]; r8[i] = r4[i] * r4[i];
    r16[i] = r8[i] * r8[i];
    rg[i]  = powi(r, col);
    api[i] = powi(PA[i], 1 + 8 * half);
  }

  // chunk-entry pole states from kernel 2
  float S[6];
#pragma unroll
  for (int i = 0; i < 6; ++i) S[i] = st[(long long)wave * 6 + i];

  for (int blk = 0; blk < NBLK; ++blk) {
    const long long b0 = chunkBase + (long long)blk * BLK_SAMP;
    __builtin_prefetch(w + b0 + BLK_SAMP + col * 16, 0, 1);   // global_prefetch

    // ---- load B (W tile) in WMMA 4x16 f32 layout: lane<16 rows {4j,4j+1},
    //      lane>=16 rows {4j+2,4j+3}, column N = lane&15. W[s,g]=w[b0+16g+s].
    v2f Bm[4];
#pragma unroll
    for (int j = 0; j < 4; ++j) {
      const float* ptr = w + b0 + col * 16 + 4 * j + 2 * half;  // 8B aligned
      Bm[j].x = ptr[0];
      Bm[j].y = ptr[1];
    }
    // contiguous copy of this lane's 8 output samples (D rows t = r + 8*half)
    const long long gfirst = b0 + col * 16 + 8 * half;
    const float4 wv0 = *reinterpret_cast<const float4*>(w + gfirst);
    const float4 wv1 = *reinterpret_cast<const float4*>(w + gfirst + 4);
    float wc[8] = {wv0.x, wv0.y, wv0.z, wv0.w, wv1.x, wv1.y, wv1.z, wv1.w};
    const long long tloc = gfirst - chanStart;                // within-channel idx
    float wpm1 = w[(tloc > 0) ? (gfirst - 1) : gfirst];       // one-sample delay
    wpm1 = (tloc > 0) ? wpm1 : 0.f;                           // b6 init = 0

    // ---- intra-column pole sum: acc = P @ W  (16x16x16 via 4x K=4 WMMAs)
    v8f acc = {0.f, 0.f, 0.f, 0.f, 0.f, 0.f, 0.f, 0.f};
#pragma unroll
    for (int j = 0; j < 4; ++j)
      acc = __builtin_amdgcn_wmma_f32_16x16x4_f32(false, Amat[j], false, Bm[j],
                                                  (short)0, acc, false, false);

    // ---- column dots d_i[g] (each lane has half its column -> xor16 combine)
    float wvp[8] = {Bm[0].x, Bm[0].y, Bm[1].x, Bm[1].y,
                    Bm[2].x, Bm[2].y, Bm[3].x, Bm[3].y};
    float d[6];
#pragma unroll
    for (int i = 0; i < 6; ++i) {
      float part = 0.f;
#pragma unroll
      for (int p = 0; p < 8; ++p) part = fmaf(wvp[p], cq[i][p], part);
      d[i] = part + __shfl_xor(part, 16, 32);
    }

    // ---- decaying inclusive scan across 16 columns:
    //      y[g] = sum_{g'<=g} r^{g-g'} d[g']   (both halves run identically)
    float y[6], sg[6], ylast[6];
#pragma unroll
    for (int i = 0; i < 6; ++i) {
      y[i] = d[i];
      float t1 = __shfl(y[i], (col >= 1) ? col - 1 : 0, 16);
      y[i] += (col >= 1) ? r1[i] * t1 : 0.f;
      float t2 = __shfl(y[i], (col >= 2) ? col - 2 : 0, 16);
      y[i] += (col >= 2) ? r2[i] * t2 : 0.f;
      float t4 = __shfl(y[i], (col >= 4) ? col - 4 : 0, 16);
      y[i] += (col >= 4) ? r4[i] * t4 : 0.f;
      float t8 = __shfl(y[i], (col >= 8) ? col - 8 : 0, 16);
      y[i] += (col >= 8) ? r8[i] * t8 : 0.f;
      // exclusive form: state entering column g = r^g * S + y[g-1]
      float yp = __shfl(y[i], (col >= 1) ? col - 1 : 0, 16);
      sg[i]   = fmaf(rg[i], S[i], (col >= 1) ? yp : 0.f);
      ylast[i] = __shfl(y[i], 15, 16);
    }

    // ---- assemble outputs: carry term a_i^{t+1}*sg_i + b6 + direct, scaled
    float res[8];
    float fac[6];
#pragma unroll
    for (int i = 0; i < 6; ++i) fac[i] = api[i];              // a^{t0+1}
#pragma unroll
    for (int r = 0; r < 8; ++r) {
      float carry = 0.f;
#pragma unroll
      for (int i = 0; i < 6; ++i) { carry = fmaf(fac[i], sg[i], carry); fac[i] *= PA[i]; }
      float wprev = (r == 0) ? wpm1 : wc[r - 1];
      res[r] = OUT_GAIN * (acc[r] + carry + B6_GAIN * wprev + DIRECT_G * wc[r]);
    }
    // ---- advance chunk states past this block
#pragma unroll
    for (int i = 0; i < 6; ++i) S[i] = fmaf(r16[i], S[i], ylast[i]);

    float4 o0 = {res[0], res[1], res[2], res[3]};
    float4 o1 = {res[4], res[5], res[6], res[7]};
    *reinterpret_cast<float4*>(out + gfirst)     = o0;
    *reinterpret_cast<float4*>(out + gfirst + 4) = o1;
  }
}

// ---------------------------------------------------------------------------
extern "C" void kernel_launch(void* const* d_in, const int* in_sizes, int n_in,
                              void* d_out, int out_size, void* d_ws, size_t ws_size,
                              hipStream_t stream) {
  const float* w = (const float*)d_in[0];
  float* out = (float*)d_out;
  const int total   = in_sizes[0];          // B * L
  const int nchunks = total / CHUNK;        // 4096 for B=256
  const int nchan   = total / L_TOTAL;      // 256

  float* f  = (float*)d_ws;                 // [nchunks][6] chunk transfers
  float* st = f + (size_t)nchunks * 6;      // [nchunks][6] chunk-entry states

  const int wavesPerBlock = 256 / 32;
  const int g1 = (nchunks + wavesPerBlock - 1) / wavesPerBlock;

  k1_chunk_reduce<<<g1, 256, 0, stream>>>(w, f, nchunks);
  k2_scan<<<(nchan + 255) / 256, 256, 0, stream>>>(f, st, nchan);
  k3_pink<<<g1, 256, 0, stream>>>(w, st, out, nchunks);
}